// DTM_PolyaGamma_15358803050960
// MI455X (gfx1250) — compile-verified
//
#include <hip/hip_runtime.h>

// ---------------------------------------------------------------------------
// DTM initialize(): pure scatter-add. Memory/atomic-bound, no matrix math.
//   out layout (floats, concatenated in reference return order):
//     flatCDK [50000,100]      @ 0            (40 MB w/ eta: keep L2-resident)
//     CWK     [10,50000,100]   @ 5,000,000    (200 MB: stream, NT)
//     CK      [10,100]         @ 55,000,000   (hot: LDS-privatized)
//     flat_eta[50000,100]      @ 55,001,000
//   total 60,001,000 floats (240 MB).
// Strategy:
//   fill phase: RT 128-bit stores for CDK/CK/eta (stay in 192 MB L2 so the
//               ~1-atomic-per-slot RMWs hit in L2), NT stores for CWK.
//   scatter:    grid-stride, 4 words/thread via b128 loads;
//               CDK/CWK/eta -> no-return global_atomic_add_f32;
//               CK (1000 hot slots) -> ds_add_f32 LDS privatization,
//               one low-contention global flush per block.
// ---------------------------------------------------------------------------

typedef int   __attribute__((ext_vector_type(4))) i32x4;
typedef float __attribute__((ext_vector_type(4))) f32x4;

#define ALL_WORD  5000000
#define NGROUPS   (ALL_WORD / 4)
#define K_        100
#define T_        10
#define V_        50000
#define NDOC_     50000

#define OFF_CDK   0
#define OFF_CWK   (NDOC_ * K_)             //  5,000,000
#define OFF_CK    (OFF_CWK + T_ * V_ * K_) // 55,000,000
#define OFF_ETA   (OFF_CK + T_ * K_)       // 55,001,000
#define OUT_N     (OFF_ETA + NDOC_ * K_)   // 60,001,000
#define N_TAIL    (OUT_N - OFF_CK)         //  5,001,000 (CK + eta, 4-divisible)

// Regular-temporal zero fill: lines stay L2-resident for the atomics phase.
__global__ __launch_bounds__(256) void zero_rt(f32x4* __restrict__ p, int n4) {
  int i = blockIdx.x * blockDim.x + threadIdx.x;
  int s = gridDim.x * blockDim.x;
  f32x4 z = {0.0f, 0.0f, 0.0f, 0.0f};
  for (; i < n4; i += s) p[i] = z;
}

// Non-temporal zero fill for the 200 MB CWK region (can't fit in L2 anyway;
// don't let it evict the 40 MB CDK/eta working set).
__global__ __launch_bounds__(256) void zero_nt(f32x4* __restrict__ p, int n4) {
  int i = blockIdx.x * blockDim.x + threadIdx.x;
  int s = gridDim.x * blockDim.x;
  f32x4 z = {0.0f, 0.0f, 0.0f, 0.0f};
  for (; i < n4; i += s) __builtin_nontemporal_store(z, p + i);
}

__global__ __launch_bounds__(256) void dtm_scatter(
    const i32x4* __restrict__ t4,   // time_ind_per_word
    const i32x4* __restrict__ d4,   // doc_indexes
    const i32x4* __restrict__ w4,   // flatW
    const i32x4* __restrict__ z4,   // flatZ
    const f32x4* __restrict__ n4,   // N_per_word
    float* __restrict__ out) {
  // Per-workgroup private CK accumulator: 10*100 floats = 4 KB of 320 KB LDS.
  __shared__ float ckLds[T_ * K_];
  for (int i = threadIdx.x; i < T_ * K_; i += blockDim.x) ckLds[i] = 0.0f;
  __syncthreads();

  int g  = blockIdx.x * blockDim.x + threadIdx.x;
  int gs = gridDim.x * blockDim.x;

  for (int i = g; i < NGROUPS; i += gs) {
    // Speculative prefetch of the next grid-stride chunk (global_prefetch_b8).
    __builtin_prefetch((const void*)(t4 + i + gs), 0, 0);
    __builtin_prefetch((const void*)(d4 + i + gs), 0, 0);
    __builtin_prefetch((const void*)(w4 + i + gs), 0, 0);
    __builtin_prefetch((const void*)(z4 + i + gs), 0, 0);
    __builtin_prefetch((const void*)(n4 + i + gs), 0, 0);

    i32x4 t = t4[i];
    i32x4 d = d4[i];
    i32x4 w = w4[i];
    i32x4 z = z4[i];
    f32x4 n = n4[i];

#pragma unroll
    for (int j = 0; j < 4; ++j) {
      int   tt = t[j], dd = d[j], ww = w[j], zz = z[j];
      float eta = 1.5f / (n[j] + 50.0f);  // (1+alpha)/(N + K*alpha), alpha=0.5

      // ~1 add per slot on average -> direct no-return global fadd is optimal.
      unsafeAtomicAdd(out + (OFF_CDK + dd * K_ + zz), 1.0f);
      unsafeAtomicAdd(out + (OFF_CWK + (tt * V_ + ww) * K_ + zz), 1.0f);
      unsafeAtomicAdd(out + (OFF_ETA + dd * K_ + zz), eta);

      // 5M adds over 1000 slots -> privatize in LDS (ds_add_f32).
      atomicAdd(&ckLds[tt * K_ + zz], 1.0f);
    }
  }

  __syncthreads();
  // Flush block-private CK: contention per slot drops from 5000 to gridDim.x.
  for (int i = threadIdx.x; i < T_ * K_; i += blockDim.x) {
    float v = ckLds[i];
    if (v != 0.0f) unsafeAtomicAdd(out + OFF_CK + i, v);
  }
}

extern "C" void kernel_launch(void* const* d_in, const int* in_sizes, int n_in,
                              void* d_out, int out_size, void* d_ws, size_t ws_size,
                              hipStream_t stream) {
  const i32x4* t4 = (const i32x4*)d_in[0];
  const i32x4* d4 = (const i32x4*)d_in[1];
  const i32x4* w4 = (const i32x4*)d_in[2];
  const i32x4* z4 = (const i32x4*)d_in[3];
  const f32x4* n4 = (const f32x4*)d_in[4];
  float* out = (float*)d_out;

  // Fill phase (must precede atomics; same stream => ordered).
  //  - CDK [0, 5M): RT stores, stays L2-resident for its 5M random atomics.
  zero_rt<<<1024, 256, 0, stream>>>((f32x4*)(out + OFF_CDK), OFF_CWK / 4);
  //  - CWK [5M, 55M): NT stores, streaming region.
  zero_nt<<<4096, 256, 0, stream>>>((f32x4*)(out + OFF_CWK), (T_ * V_ * K_) / 4);
  //  - CK + eta [55M, 60,001,000): RT stores.
  zero_rt<<<1024, 256, 0, stream>>>((f32x4*)(out + OFF_CK), N_TAIL / 4);

  // Scatter. 1024 blocks x 256 thr = 262,144 threads (~8192 waves), ~19 words
  // each: saturates the L2 atomic pipes while keeping the CK flush at ~1M
  // low-contention atomics.
  dtm_scatter<<<1024, 256, 0, stream>>>(t4, d4, w4, z4, n4, out);
}